// remove_color_26792005992696
// MI455X (gfx1250) — compile-verified
//
#include <hip/hip_runtime.h>
#include <stdint.h>

#define AS1 __attribute__((address_space(1)))
#define AS3 __attribute__((address_space(3)))

#if defined(__has_builtin) && __has_builtin(__builtin_amdgcn_global_load_async_to_lds_b32)
#define HAVE_ASYNC_LDS 1
#else
#define HAVE_ASYNC_LDS 0
#endif

namespace {
constexpr int C = 3, H = 256, W = 256, R = 2, K = 5;     // L = K*K = 25 shifts
constexpr int TW = 32, TH = 8;                            // pixel tile per block
constexpr int LW = TW + 2 * R;                            // 36
constexpr int LH = TH + 2 * R;                            // 12
constexpr int TILE_ELEMS = LH * LW * C;                   // 1296 floats = 5184 B LDS
constexpr int NTHREADS = 256;                             // 8 waves (wave32)
}

// out[(i*25+j), y, x] = both-in-bounds ? sum_c |img[c,y+dyi,x+dxi]-img[c,y+dyj,x+dxj]| : 0
// blockIdx.z == dy_i (ady): each block writes 125 of the 625 planes for its tile.
__global__ __launch_bounds__(NTHREADS) void cost_volume(const float* __restrict__ img,
                                                        float* __restrict__ out) {
  __shared__ float tile[TILE_ELEMS];  // layout [LH][LW][C] -> pixel's 3 ch contiguous (b96)

  const int tid = threadIdx.x;
  const int bx = blockIdx.x, by = blockIdx.y;
  const int x0 = bx * TW - R;
  const int y0 = by * TH - R;

  // ---- Stage tile + halo into LDS (edge-clamped; validity handled at compute) ----
  for (int e = tid; e < TILE_ELEMS; e += NTHREADS) {
    const int c  = e % C;
    const int t  = e / C;
    const int ly = t / LW;
    const int lx = t - ly * LW;
    int gy = y0 + ly; gy = gy < 0 ? 0 : (gy > H - 1 ? H - 1 : gy);
    int gx = x0 + lx; gx = gx < 0 ? 0 : (gx > W - 1 ? W - 1 : gx);
    const float* src = img + (c * (H * W) + gy * W + gx);
#if HAVE_ASYNC_LDS
    // CDNA5 async global->LDS copy (tracked by ASYNCcnt, no VGPR staging)
    __builtin_amdgcn_global_load_async_to_lds_b32(
        (AS1 int*)(uintptr_t)src,
        (AS3 int*)(uint32_t)(uintptr_t)&tile[e],
        /*imm offset*/ 0, /*cpol*/ 0);
#else
    tile[e] = *src;
#endif
  }
#if HAVE_ASYNC_LDS
#if defined(__has_builtin) && __has_builtin(__builtin_amdgcn_s_wait_asynccnt)
  __builtin_amdgcn_s_wait_asynccnt(0);
#else
  asm volatile("s_wait_asynccnt 0" ::: "memory");
#endif
#endif
  __syncthreads();

  // ---- Compute 125 planes (ady fixed per block via blockIdx.z) ----
  const int tx = tid & (TW - 1);
  const int ty = tid >> 5;
  const int x = bx * TW + tx;
  const int y = by * TH + ty;
  const int ady = blockIdx.z;  // 0..4  => dy_i = ady - R

  const bool aok_y = (unsigned)(y + ady - R) < (unsigned)H;
  float* outp = out + (size_t)(ady * 125) * (H * W) + (size_t)(y * W + x);

  for (int adx = 0; adx < K; ++adx) {
    const bool aok = aok_y && ((unsigned)(x + adx - R) < (unsigned)W);
    const float* ap = &tile[((ty + ady) * LW + (tx + adx)) * C];
    const float a0 = ap[0], a1 = ap[1], a2 = ap[2];
    for (int bdy = 0; bdy < K; ++bdy) {
      const bool bok_y = (unsigned)(y + bdy - R) < (unsigned)H;
      const int brow = (ty + bdy) * LW;
#pragma unroll
      for (int bdx = 0; bdx < K; ++bdx) {
        const bool bok = bok_y && ((unsigned)(x + bdx - R) < (unsigned)W);
        const float* bp = &tile[(brow + (tx + bdx)) * C];
        const float s = fabsf(a0 - bp[0]) + fabsf(a1 - bp[1]) + fabsf(a2 - bp[2]);
        const float v = (aok && bok) ? s : 0.0f;
        const int p = (adx * K + bdy) * K + bdx;  // local plane 0..124
        // 164 MB write stream, never re-read soon -> non-temporal store
        __builtin_nontemporal_store(v, outp + (size_t)p * (H * W));
      }
    }
  }
}

extern "C" void kernel_launch(void* const* d_in, const int* in_sizes, int n_in,
                              void* d_out, int out_size, void* d_ws, size_t ws_size,
                              hipStream_t stream) {
  (void)in_sizes; (void)n_in; (void)d_ws; (void)ws_size; (void)out_size;
  const float* img = (const float*)d_in[0];   // (3, 256, 256) f32
  float* out = (float*)d_out;                 // (625, 256, 256) f32

  dim3 grid(W / TW, H / TH, K);               // (8, 32, 5) = 1280 blocks
  dim3 block(NTHREADS, 1, 1);                 // 8 waves of 32
  cost_volume<<<grid, block, 0, stream>>>(img, out);
}